// GIGM_5952824672776
// MI455X (gfx1250) — compile-verified
//
#include <hip/hip_runtime.h>
#include <hip/hip_bf16.h>
#include <math.h>

// Problem constants (match reference)
#define BB 16
#define TT 64
#define AA 8
#define NN 8
#define FF 16
#define MM 8
#define FE 16
#define NS 10
#define TP1 65
#define EMB 18

typedef __attribute__((ext_vector_type(2))) float v2f;
typedef __attribute__((ext_vector_type(8))) float v8f;

// -------- restricted value on an 8-node coalition (bitmask connected components) -------
__device__ __forceinline__ float restricted_value(unsigned C,
                                                  const unsigned* adjb,
                                                  const float adjf[NN][NN],
                                                  const float* q) {
    // reachability closure: reach[n] = component mask of n within coalition C
    unsigned reach[NN];
#pragma unroll
    for (int n = 0; n < NN; ++n)
        reach[n] = ((C >> n) & 1u) ? ((adjb[n] & C) | (1u << n)) : (1u << n);
#pragma unroll
    for (int it = 0; it < 3; ++it) {   // R -> R^2 -> R^4 -> R^8 covers all paths (diam <= 7)
#pragma unroll
        for (int n = 0; n < NN; ++n) {
            unsigned r = reach[n], acc = r;
#pragma unroll
            for (int m = 0; m < NN; ++m)
                if ((r >> m) & 1u) acc |= reach[m];
            reach[n] = acc;
        }
    }
    float rv = 0.f;
#pragma unroll
    for (int n = 0; n < NN; ++n) {
        if (!((C >> n) & 1u)) continue;
        unsigned comp = reach[n];
        // n is the representative iff it is the minimum index in its component
        if ((comp & ((1u << n) - 1u)) != 0u) continue;
        int size = __popc(comp);
        float util = 0.f;
#pragma unroll
        for (int m = 0; m < NN; ++m)
            if ((comp >> m) & 1u) util += q[m];
        float dens;
        if (size > 1) {
            float e = 0.f;
#pragma unroll
            for (int k = 0; k < NN; ++k) {
                if (!((comp >> k) & 1u)) continue;
#pragma unroll
                for (int m = 0; m < NN; ++m)
                    if ((comp >> m) & 1u) e += adjf[k][m];
            }
            dens = e / (float)(size * (size - 1));
        } else {
            dens = 1.f;
        }
        rv += dens * util;
    }
    return rv;
}

// -------- stage 1: per (b,t,a) -> mu, alive ------------------------------------------
__global__ void gigm_stage1(const float* __restrict__ all_node,
                            const float* __restrict__ all_adj,
                            const float* __restrict__ qvals,
                            const float* __restrict__ permr,
                            float* __restrict__ mu_out,
                            float* __restrict__ alive_out) {
    int idx = blockIdx.x * blockDim.x + threadIdx.x;
    if (idx >= BB * TT * AA) return;
    int a  = idx % AA;
    int bt = idx / AA;
    int t  = bt % TT;
    int b  = bt / TT;

    const float* node = all_node + ((((size_t)b * TP1 + t) * AA + a) * NN) * FF;
    unsigned vis = 0;
#pragma unroll
    for (int n = 0; n < NN; ++n)
        if (node[n * FF] > 0.f) vis |= (1u << n);
    float alive = (vis != 0u) ? 1.f : 0.f;

    const float* adjp = all_adj + ((((size_t)b * TP1 + t) * AA + a) * NN) * NN;
    float adjf[NN][NN];
    unsigned adjb[NN];
#pragma unroll
    for (int n = 0; n < NN; ++n) {
        unsigned bm = 0;
#pragma unroll
        for (int m = 0; m < NN; ++m) {
            float v = adjp[n * NN + m];
            adjf[n][m] = v;
            if (v > 0.f) bm |= (1u << m);
        }
        adjb[n] = bm;
    }

    float q[NN];
    const float* qp = qvals + ((size_t)b * TT + t) * AA;
#pragma unroll
    for (int n = 0; n < NN; ++n) q[n] = qp[n];

    const float* pr = permr + ((((size_t)b * TT + t) * AA + a) * NS) * NN;
    float acc = 0.f;
    for (int s = 0; s < NS; ++s) {
        float rank[NN];
#pragma unroll
        for (int n = 0; n < NN; ++n) rank[n] = pr[s * NN + n];
        float sr = rank[0];
#pragma unroll
        for (int n = 1; n < NN; ++n) sr = (a == n) ? rank[n] : sr;

        unsigned prev = 0;
#pragma unroll
        for (int n = 0; n < NN; ++n)
            if (((vis >> n) & 1u) && (rank[n] < sr)) prev |= (1u << n);
        unsigned withm = prev | (1u << a);

        acc += restricted_value(withm, adjb, adjf, q) -
               restricted_value(prev,  adjb, adjf, q);
    }
    float mc = acc / (float)NS;
    float mu = ((vis >> a) & 1u) ? mc : 0.f;
    mu_out[idx]    = mu;
    alive_out[idx] = alive;
}

// -------- enemy-graph embedding (18-dim) ---------------------------------------------
__device__ __forceinline__ void enemy_embed(const float* __restrict__ node,  // [MM][FE]
                                            const float* __restrict__ adj,   // [MM][MM]
                                            float* __restrict__ emb) {
    float m[MM];
    float cnt = 0.f;
#pragma unroll
    for (int j = 0; j < MM; ++j) { m[j] = node[j * FE]; cnt += m[j]; }
    cnt = fmaxf(cnt, 1.f);
#pragma unroll
    for (int f = 0; f < FE; ++f) {
        float s = 0.f;
#pragma unroll
        for (int j = 0; j < MM; ++j) s += node[j * FE + f] * m[j];
        emb[f] = s / cnt;
    }
    float deg[MM];
#pragma unroll
    for (int j = 0; j < MM; ++j) {
        float s = 0.f;
#pragma unroll
        for (int k = 0; k < MM; ++k) s += adj[j * MM + k];
        deg[j] = s / (float)(MM - 1);
    }
    float md = 0.f;
#pragma unroll
    for (int j = 0; j < MM; ++j) md += deg[j] * m[j];
    md /= cnt;
    float vd = 0.f;
#pragma unroll
    for (int j = 0; j < MM; ++j) {
        float d = deg[j] - md;
        vd += d * d * m[j];
    }
    vd /= cnt;
    emb[16] = md;
    emb[17] = vd;
}

__device__ __forceinline__ void softmax18(const float* __restrict__ e, float* __restrict__ p) {
    float mx = e[0];
#pragma unroll
    for (int i = 1; i < EMB; ++i) mx = fmaxf(mx, e[i]);
    float s = 0.f;
#pragma unroll
    for (int i = 0; i < EMB; ++i) { float v = expf(e[i] - mx); p[i] = v; s += v; }
    float inv = 1.f / s;
#pragma unroll
    for (int i = 0; i < EMB; ++i) p[i] *= inv;
}

// -------- stage 2: per (b,t,a) -> inc -------------------------------------------------
__global__ void gigm_stage2(const float* __restrict__ ene_node,
                            const float* __restrict__ ene_adj,
                            const float* __restrict__ maskp,
                            const float* __restrict__ mu_ws,
                            const float* __restrict__ alive_ws,
                            float* __restrict__ inc_out) {
    int idx = blockIdx.x * blockDim.x + threadIdx.x;
    if (idx >= BB * TT * AA) return;
    int a  = idx % AA;
    int bt = idx / AA;
    int t  = bt % TT;
    int b  = bt / TT;

    // intrinsic: exp(clip(mu - mean_mu)) * alive
    const float* mub = mu_ws    + (size_t)bt * AA;
    const float* alb = alive_ws + (size_t)bt * AA;
    float sum_mu = 0.f, sum_al = 0.f;
#pragma unroll
    for (int j = 0; j < AA; ++j) { float al = alb[j]; sum_mu += mub[j] * al; sum_al += al; }
    float mean_mu = sum_mu / fmaxf(sum_al, 1.f);
    float x = mu_ws[idx] - mean_mu;
    x = fminf(fmaxf(x, -5.f), 5.f);
    float intrinsic = expf(x) * alive_ws[idx];

    // extrinsic: KL(softmax(emb_{t+1}) || softmax(emb_t)) * alive_e
    const float* nod_t   = ene_node + ((((size_t)b * TP1 + t    ) * AA + a) * MM) * FE;
    const float* nod_tp1 = ene_node + ((((size_t)b * TP1 + t + 1) * AA + a) * MM) * FE;
    const float* adj_t   = ene_adj  + ((((size_t)b * TP1 + t    ) * AA + a) * MM) * MM;
    const float* adj_tp1 = ene_adj  + ((((size_t)b * TP1 + t + 1) * AA + a) * MM) * MM;

    float e_t[EMB], e_tp1[EMB], p[EMB], qd[EMB];
    enemy_embed(nod_t,   adj_t,   e_t);
    enemy_embed(nod_tp1, adj_tp1, e_tp1);
    softmax18(e_tp1, p);
    softmax18(e_t,   qd);
    float kl = 0.f;
#pragma unroll
    for (int i = 0; i < EMB; ++i)
        kl += p[i] * (logf(p[i] + 1e-8f) - logf(qd[i] + 1e-8f));

    float mx = nod_t[0];
#pragma unroll
    for (int j = 1; j < MM; ++j) mx = fmaxf(mx, nod_t[j * FE]);
    float alive_e = (mx > 0.f) ? 1.f : 0.f;
    float extrinsic = kl * alive_e;

    float inc = (2.0f * intrinsic + 1.0f * extrinsic) * 0.05f * maskp[bt];
    inc_out[idx] = inc;
}

// -------- stage 3: team = sum over agents, via V_WMMA_F32_16X16X4_F32 -----------------
// One 32-lane wave per block handles 16 (b,t) rows. A(16x4) carries inc for 4 agents
// (f32 A layout: lanes 0-15 -> K={0,1}, lanes 16-31 -> K={2,3}); B = ones (layout
// invariant), two chained WMMAs accumulate agents 0-3 then 4-7. D layout: VGPR v,
// lanes 0-15 -> row v, lanes 16-31 -> row v+8 (replicated across N), so lanes 0 and 16
// hold all 16 row sums.
__global__ void gigm_team_wmma(const float* __restrict__ inc,
                               float* __restrict__ team) {
    int lane = threadIdx.x;            // 0..31 (block = exactly one wave, EXEC all 1s)
    int tile = blockIdx.x;             // 0..63 ; rows tile*16 .. tile*16+15
    int row  = tile * 16 + (lane & 15);
    int kb   = (lane < 16) ? 0 : 2;
    const float* rp = inc + (size_t)row * AA;

    v2f a0, a1, bones;
    a0.x = rp[kb + 0]; a0.y = rp[kb + 1];   // agents 0..3 as K=0..3
    a1.x = rp[kb + 4]; a1.y = rp[kb + 5];   // agents 4..7 as K=0..3
    bones.x = 1.0f;    bones.y = 1.0f;

    v8f c = {};
    c = __builtin_amdgcn_wmma_f32_16x16x4_f32(false, a0, false, bones, (short)0, c, false, false);
    c = __builtin_amdgcn_wmma_f32_16x16x4_f32(false, a1, false, bones, (short)0, c, false, false);

    if (lane == 0) {
#pragma unroll
        for (int v = 0; v < 8; ++v) team[tile * 16 + v] = c[v];
    } else if (lane == 16) {
#pragma unroll
        for (int v = 0; v < 8; ++v) team[tile * 16 + 8 + v] = c[v];
    }
}

extern "C" void kernel_launch(void* const* d_in, const int* in_sizes, int n_in,
                              void* d_out, int out_size, void* d_ws, size_t ws_size,
                              hipStream_t stream) {
    const float* all_node = (const float*)d_in[0];
    const float* all_adj  = (const float*)d_in[1];
    const float* ene_node = (const float*)d_in[2];
    const float* ene_adj  = (const float*)d_in[3];
    const float* qvals    = (const float*)d_in[4];
    const float* maskp    = (const float*)d_in[5];
    const float* permr    = (const float*)d_in[6];

    float* out      = (float*)d_out;            // [0,8192): inc ; [8192,9216): team
    float* mu_ws    = (float*)d_ws;             // 8192 floats
    float* alive_ws = mu_ws + BB * TT * AA;     // 8192 floats

    const int nBTA = BB * TT * AA;              // 8192
    gigm_stage1<<<(nBTA + 255) / 256, 256, 0, stream>>>(all_node, all_adj, qvals, permr,
                                                        mu_ws, alive_ws);
    gigm_stage2<<<(nBTA + 255) / 256, 256, 0, stream>>>(ene_node, ene_adj, maskp,
                                                        mu_ws, alive_ws, out);
    gigm_team_wmma<<<(BB * TT) / 16, 32, 0, stream>>>(out, out + nBTA);
}